// SumPooling_34531537060556
// MI455X (gfx1250) — compile-verified
//
#include <hip/hip_runtime.h>
#include <stdint.h>

// Sorted segment-sum: out[b, :] = sum over rows i with batch[i] == b of x[i, :]
// x: (N, 128) f32, batch: (N,) i32 sorted ascending, out: (B, 128) f32.
//
// Pass 1 (O(N)): build CSR offsets row_ptr[0..B] in d_ws.
// Pass 2: one wave32 per segment; lane l owns columns [4l, 4l+4) so a row is
// exactly one wave-wide b128 transaction (32 x 16B = 512B). Bulk rows are
// streamed with GLOBAL_LOAD_ASYNC_TO_LDS_B128 (ASYNCcnt-tracked, double-
// buffered 4-row tiles in LDS), reduced via ds_load_b128 into registers.
// No atomics, fixed summation order -> deterministic.

typedef float v4f __attribute__((ext_vector_type(4)));

#define WAVES_PER_BLOCK 8
#define TILE 4  // rows per async tile (TILE*512B = 2KB per buffer)

// ---------------- Pass 1: CSR row pointers ----------------
__global__ __launch_bounds__(256) void build_rowptr_kernel(
    const int* __restrict__ batch, int* __restrict__ rowptr, int N, int B) {
  const int i = blockIdx.x * blockDim.x + threadIdx.x;
  if (i >= N) return;
  const int b = batch[i];
  const int prev = (i == 0) ? -1 : batch[i - 1];
  for (int s = prev + 1; s <= b; ++s) rowptr[s] = i;   // covers empty segments
  if (i == N - 1) {
    for (int s = b + 1; s <= B; ++s) rowptr[s] = N;    // trailing empties + end
  }
}

// ---------------- async helpers (CDNA5 ISA) ----------------
__device__ __forceinline__ void async_b128_to_lds(const float* gsrc,
                                                  const v4f* lds_dst) {
  // Low 32 bits of a generic pointer into LDS == workgroup-relative LDS offset.
  uint32_t lds_off = (uint32_t)(uintptr_t)lds_dst;
  uint64_t gaddr   = (uint64_t)(uintptr_t)gsrc;
  asm volatile("global_load_async_to_lds_b128 %0, %1, off"
               :: "v"(lds_off), "v"(gaddr)
               : "memory");
}

__device__ __forceinline__ void wait_asynccnt_le_tile() {
  asm volatile("s_wait_asynccnt %0" :: "n"(TILE) : "memory");
}
__device__ __forceinline__ void wait_asynccnt_0() {
  asm volatile("s_wait_asynccnt 0" ::: "memory");
}
__device__ __forceinline__ void wait_dscnt_0() {
  asm volatile("s_wait_dscnt 0" ::: "memory");
}

// Fallback (ws too small): plain lower_bound over the sorted batch array.
__device__ __forceinline__ int lower_bound_i32(const int* __restrict__ a,
                                               int n, int key) {
  int lo = 0, hi = n;
  while (lo < hi) {
    int mid = (lo + hi) >> 1;
    if (a[mid] < key) lo = mid + 1; else hi = mid;
  }
  return lo;
}

// ---------------- Pass 2: segment reduction ----------------
__global__ __launch_bounds__(WAVES_PER_BLOCK * 32) void segsum_kernel(
    const float* __restrict__ x,
    const int* __restrict__ batch,
    const int* __restrict__ rowptr,   // nullptr -> use binary search
    float* __restrict__ out,
    int N, int B) {
  __shared__ v4f lds[WAVES_PER_BLOCK][2][TILE][32];

  const int wave = threadIdx.x >> 5;    // wave32
  const int lane = threadIdx.x & 31;
  const int seg  = blockIdx.x * WAVES_PER_BLOCK + wave;
  if (seg >= B) return;

  int start, end;
  if (rowptr != nullptr) {
    start = rowptr[seg];
    end   = rowptr[seg + 1];
  } else {
    start = lower_bound_i32(batch, N, seg);
    end   = lower_bound_i32(batch, N, seg + 1);
  }

  v4f a0 = {0.f, 0.f, 0.f, 0.f};
  v4f a1 = {0.f, 0.f, 0.f, 0.f};

  int r = start;
  const int nrows  = end - start;
  const int ntiles = nrows / TILE;

  if (ntiles > 0) {
    const float* gbase = x + (size_t)lane * 4;
    // prologue: issue tile 0 into buffer 0
    {
      const float* g = gbase + (size_t)r * 128;
      #pragma unroll
      for (int i = 0; i < TILE; ++i)
        async_b128_to_lds(g + (size_t)i * 128, &lds[wave][0][i][lane]);
    }
    for (int t = 1; t < ntiles; ++t) {
      // issue tile t into buffer t&1
      const float* g = gbase + (size_t)(r + t * TILE) * 128;
      #pragma unroll
      for (int i = 0; i < TILE; ++i)
        async_b128_to_lds(g + (size_t)i * 128, &lds[wave][t & 1][i][lane]);
      // previous tile's 4 async loads are done once only TILE remain in flight
      wait_asynccnt_le_tile();
      const int pb = (t - 1) & 1;
      v4f v0 = lds[wave][pb][0][lane];
      v4f v1 = lds[wave][pb][1][lane];
      v4f v2 = lds[wave][pb][2][lane];
      v4f v3 = lds[wave][pb][3][lane];
      a0 += v0; a1 += v1; a0 += v2; a1 += v3;
      wait_dscnt_0();   // DS reads retired before this buffer is refilled
    }
    // epilogue: drain and consume the last tile
    wait_asynccnt_0();
    {
      const int pb = (ntiles - 1) & 1;
      v4f v0 = lds[wave][pb][0][lane];
      v4f v1 = lds[wave][pb][1][lane];
      v4f v2 = lds[wave][pb][2][lane];
      v4f v3 = lds[wave][pb][3][lane];
      a0 += v0; a1 += v1; a0 += v2; a1 += v3;
    }
    r += ntiles * TILE;
  }

  // tail rows (< TILE): direct nontemporal b128 loads
  for (; r < end; ++r) {
    a0 += __builtin_nontemporal_load((const v4f*)x + ((long)r << 5) + lane);
  }

  const v4f acc = a0 + a1;
  ((v4f*)out)[((long)seg << 5) + lane] = acc;   // exactly one writer per element
}

// ---------------- host launcher ----------------
extern "C" void kernel_launch(void* const* d_in, const int* in_sizes, int n_in,
                              void* d_out, int out_size, void* d_ws, size_t ws_size,
                              hipStream_t stream) {
  const float* x     = (const float*)d_in[0];
  const int*   batch = (const int*)d_in[1];
  float* out = (float*)d_out;

  const int D = 128;
  const int N = in_sizes[0] / D;        // 2,000,000
  const int B = out_size / D;           // 100,000

  int* rowptr = nullptr;
  if (ws_size >= (size_t)(B + 1) * sizeof(int)) {
    rowptr = (int*)d_ws;
    const int threads = 256;
    build_rowptr_kernel<<<(N + threads - 1) / threads, threads, 0, stream>>>(
        batch, rowptr, N, B);
  }

  const int blocks = (B + WAVES_PER_BLOCK - 1) / WAVES_PER_BLOCK;
  segsum_kernel<<<blocks, WAVES_PER_BLOCK * 32, 0, stream>>>(
      x, batch, rowptr, out, N, B);
}